// Sampler_89850715833153
// MI455X (gfx1250) — compile-verified
//
#include <hip/hip_runtime.h>
#include <hip/hip_bf16.h>
#include <math.h>

// ---------------------------------------------------------------------------
// Sampler pipeline for gfx1250 (MI455X).  Bandwidth-bound; no matmul -> no
// WMMA (it would be artificial).  Exploits wave32 shuffles/ballots,
// deterministic LDS integer atomics (ds_add_u32/u64), split s_wait_*
// counters, 16B-per-lane global accesses, and global_prefetch_b8.
//
// Key structure: all three mask conditions (min-p/top-a threshold, top-p,
// top-k) are suffixes of the stable sort order (value desc, index asc), so
// the kept set is exactly the top-K* elements.  K* and the K*-th key are
// found with an exact MSD radix descent over order-preserving uint32 keys
// using 2^40 fixed-point uint64 exp-sums -> bit-deterministic, no sort.
// ---------------------------------------------------------------------------

static constexpr int TB = 1024;                       // analysis block
static constexpr double FPSCALE = 1099511627776.0;    // 2^40 fixed point

__device__ __forceinline__ unsigned fkey(float f) {
  // order-preserving map: f1 < f2  <=>  fkey(f1) < fkey(f2)
  unsigned b = __float_as_uint(f);
  return (b & 0x80000000u) ? ~b : (b | 0x80000000u);
}
__device__ __forceinline__ float keyf(unsigned u) {
  unsigned b = (u & 0x80000000u) ? (u & 0x7fffffffu) : ~u;
  return __uint_as_float(b);
}
__device__ __forceinline__ float neg_inf() { return __uint_as_float(0xff800000u); }

// ---------------------------------------------------------------------------
// K0: reset per-row max-key slots (d_ws is not re-poisoned between replays).
// ---------------------------------------------------------------------------
__global__ void k0_init(unsigned* __restrict__ rowmax, int B) {
  int i = blockIdx.x * 256 + threadIdx.x;
  if (i < B) rowmax[i] = 0u;
}

// ---------------------------------------------------------------------------
// K1: dense pass: out = logits / temp (penalty-free baseline; K2 fixes up
// the <=1280 penalized entries/row).  float4 path: 1024 elements per block.
// Also accumulates the per-row max key (upper bound on the final penalized
// max) via integer atomicMax.
// ---------------------------------------------------------------------------
__global__ __launch_bounds__(256) void k1_scale(const float* __restrict__ logits,
                                                const float* __restrict__ temps,
                                                float* __restrict__ out,
                                                unsigned* __restrict__ rowmax,
                                                int V) {
  const int row = blockIdx.y;
  const int c0 = blockIdx.x * 1024 + threadIdx.x * 4;
  __shared__ unsigned red[256];

  float tv = temps[row];
  tv = (tv == 0.0f) ? 1.0f : tv;
  const size_t ro = (size_t)row * V;

  unsigned key = 0u;
  if (c0 + 3 < V) {
    float4 x = *(const float4*)(logits + ro + c0);
    x.x /= tv; x.y /= tv; x.z /= tv; x.w /= tv;
    *(float4*)(out + ro + c0) = x;
    unsigned k0 = fkey(x.x), k1 = fkey(x.y), k2 = fkey(x.z), k3 = fkey(x.w);
    k0 = (k0 > k1) ? k0 : k1;
    k2 = (k2 > k3) ? k2 : k3;
    key = (k0 > k2) ? k0 : k2;
  } else {
    for (int j = 0; j < 4; ++j) {
      int c = c0 + j;
      if (c < V) {
        float v = logits[ro + c] / tv;
        out[ro + c] = v;
        unsigned k = fkey(v);
        key = (k > key) ? k : key;
      }
    }
  }
  red[threadIdx.x] = key;
  __syncthreads();
  for (int s = 128; s > 0; s >>= 1) {
    if (threadIdx.x < (unsigned)s) {
      unsigned a = red[threadIdx.x], b = red[threadIdx.x + s];
      red[threadIdx.x] = (a > b) ? a : b;
    }
    __syncthreads();
  }
  if (threadIdx.x == 0) atomicMax(&rowmax[row], red[0]);
}

// ---------------------------------------------------------------------------
// K2: sparse penalty fixup.  One 256-thread block per row.
//  - bitonic-sort the LO=256 output tokens in LDS
//  - first-of-run thread applies rep -> freq*count -> presence -> temp
//  - prompt tokens not in the output set get rep -> temp
// Duplicate writers always write identical values (benign race).  Every
// written value also ratchets rowmax, so the bound holds for any inputs.
// ---------------------------------------------------------------------------
__global__ __launch_bounds__(256) void k2_penalize(
    const float* __restrict__ logits, const int* __restrict__ prompt,
    const int* __restrict__ outtok, const float* __restrict__ pres,
    const float* __restrict__ freq, const float* __restrict__ rep,
    const float* __restrict__ temps, float* __restrict__ out,
    unsigned* __restrict__ rowmax, int V, int LP, int LO) {
  const int row = blockIdx.x;
  const int tid = threadIdx.x;
  __shared__ int O[256];

  O[tid] = outtok[(size_t)row * LO + tid];
  __syncthreads();

  // bitonic sort ascending (n = 256)
  for (int k = 2; k <= 256; k <<= 1) {
    for (int j = k >> 1; j > 0; j >>= 1) {
      int ixj = tid ^ j;
      if (ixj > tid) {
        int a = O[tid], b = O[ixj];
        bool up = ((tid & k) == 0);
        if ((up && a > b) || (!up && a < b)) { O[tid] = b; O[ixj] = a; }
      }
      __syncthreads();
    }
  }

  const float rv = rep[row], fv = freq[row], pv = pres[row];
  float tv = temps[row];
  tv = (tv == 0.0f) ? 1.0f : tv;

  // unique output tokens: rep -> freq -> presence -> temp (reference order)
  {
    int v = O[tid];
    bool first = (tid == 0) || (O[tid - 1] != v);
    if (first) {
      int end = tid + 1;
      while (end < 256 && O[end] == v) ++end;
      int cnt = end - tid;
      float l = logits[(size_t)row * V + v];
      l = (l > 0.0f) ? (l / rv) : (l * rv);
      l -= fv * (float)cnt;
      l -= pv;
      l /= tv;
      out[(size_t)row * V + v] = l;
      atomicMax(&rowmax[row], fkey(l));
    }
  }

  // prompt-only tokens: rep -> temp
  for (int p = tid; p < LP; p += 256) {
    int t = prompt[(size_t)row * LP + p];
    int lo = 0, hi = 255;
    bool found = false;
    while (lo <= hi) {
      int mid = (lo + hi) >> 1;
      int ov = O[mid];
      if (ov == t) { found = true; break; }
      if (ov < t) lo = mid + 1; else hi = mid - 1;
    }
    if (!found) {
      float l = logits[(size_t)row * V + t];
      l = (l > 0.0f) ? (l / rv) : (l * rv);
      l /= tv;
      out[(size_t)row * V + t] = l;
      atomicMax(&rowmax[row], fkey(l));
    }
  }
}

// ---------------------------------------------------------------------------
// K3: per-row selection + finalize (fused).  One TB-thread block per row.
// All full-row passes use float4 (global_load/store_b128).
//  scan 1 : level-0 histogram (count + u64 fixed-point exp-sums) + true max
//  scans 2-4 : top-p radix descent; scan 2 also counts the min-p/top-a set
//  scans 5-7 : rank descent for K*-th key — SKIPPED when top-p binds
//  scan 8 : finalize in place; exact index-ordered tie scan via wave32
//           __shfl_up scan of per-thread tie counts.
// ---------------------------------------------------------------------------
__global__ __launch_bounds__(TB) void k3_select_finalize(
    float* __restrict__ gout, const unsigned* __restrict__ rowmax,
    const float* __restrict__ top_ps, const float* __restrict__ top_as,
    const float* __restrict__ min_ps, const int* __restrict__ top_ks, int V) {
  const int row = blockIdx.x;
  float* rp = gout + (size_t)row * V;
  const float4* rp4 = (const float4*)rp;
  const int nv4 = V >> 2;          // # whole float4 elements
  const int tail = nv4 << 2;       // first scalar-tail index
  const int tid = threadIdx.x;

  __shared__ unsigned redu[TB];
  __shared__ unsigned cnt0[256];
  __shared__ unsigned long long esum0[256];
  __shared__ unsigned cntL[256];
  __shared__ unsigned long long esumL[256];
  __shared__ float s_ethr;
  __shared__ unsigned s_cthr;
  __shared__ unsigned s_pfx;
  __shared__ unsigned s_Kstar;
  __shared__ int s_needRank;
  __shared__ unsigned s_kkey, s_q;
  __shared__ unsigned wtot[32], wexcl[32];
  __shared__ unsigned s_btot;

  const float m = keyf(rowmax[row]);   // >= max(final row values): exp <= 1

  // ---- scan 1: level-0 histogram + true max key ----
  if (tid < 256) { cnt0[tid] = 0u; esum0[tid] = 0ull; }
  __syncthreads();
  unsigned km = 0u;
  for (int i = tid; i < nv4; i += TB) {
    __builtin_prefetch(rp4 + i + 4 * TB, 0, 0);
    float4 x = rp4[i];
    float xs[4] = {x.x, x.y, x.z, x.w};
#pragma unroll
    for (int j = 0; j < 4; ++j) {
      float l = xs[j];
      unsigned u = fkey(l);
      km = (u > km) ? u : km;
      float e = expf(l - m);
      unsigned long long E = (unsigned long long)((double)e * FPSCALE);
      atomicAdd(&cnt0[u >> 24], 1u);
      atomicAdd(&esum0[u >> 24], E);
    }
  }
  for (int i = tail + tid; i < V; i += TB) {
    float l = rp[i];
    unsigned u = fkey(l);
    km = (u > km) ? u : km;
    float e = expf(l - m);
    atomicAdd(&cnt0[u >> 24], 1u);
    atomicAdd(&esum0[u >> 24], (unsigned long long)((double)e * FPSCALE));
  }
  redu[tid] = km;
  __syncthreads();
  for (int s = TB / 2; s > 0; s >>= 1) {
    if (tid < s) {
      unsigned a = redu[tid], b = redu[tid + s];
      redu[tid] = (a > b) ? a : b;
    }
    __syncthreads();
  }

  // tid-0 persistent descent state
  unsigned long long P = 0ull, acc = 0ull;
  unsigned Crun = 0u, pfx = 0u;

  if (tid == 0) {
    unsigned long long zt = 0ull;                 // Z in fixed units (exact)
    for (int b = 0; b < 256; ++b) zt += esum0[b];
    if (zt == 0ull) zt = 1ull;                    // degenerate guard
    double Z = (double)zt / FPSCALE;
    double emax = (double)expf(keyf(redu[0]) - m);  // true p_top numerator
    // p >= thr  <=>  e >= max(emax*min_p, emax^2*top_a/Z)
    double ethr = fmax(emax * (double)min_ps[row],
                       emax * emax * (double)top_as[row] / Z);
    s_ethr = (float)ethr;
    s_cthr = 0u;
    P = (unsigned long long)((double)top_ps[row] * (double)zt);   // budget
    // level-0 top-p walk
    int fb = -1, lastNE = -1;
    unsigned long long accNE = 0ull; unsigned CNE = 0u;
    for (int b = 255; b >= 0; --b) {
      unsigned c = cnt0[b]; unsigned long long sb = esum0[b];
      if (c && acc + sb > P) { fb = b; break; }
      if (c) { lastNE = b; accNE = acc; CNE = Crun; }
      acc += sb; Crun += c;
    }
    if (fb < 0) { fb = lastNE; acc = accNE; Crun = CNE; }  // rounding guard
    pfx = (unsigned)fb;
    s_pfx = pfx;
  }
  __syncthreads();

  // ---- scans 2-4: top-p descent (scan 2 also counts threshold set) ----
  for (int lv = 1; lv <= 3; ++lv) {
    if (tid < 256) { cntL[tid] = 0u; esumL[tid] = 0ull; }
    __syncthreads();
    const unsigned p = s_pfx;
    const float ethr = s_ethr;
    const int hish = 32 - 8 * lv;
    const int bsh = 24 - 8 * lv;
    unsigned locthr = 0u;
    for (int i = tid; i < nv4; i += TB) {
      __builtin_prefetch(rp4 + i + 4 * TB, 0, 0);
      float4 x = rp4[i];
      float xs[4] = {x.x, x.y, x.z, x.w};
#pragma unroll
      for (int j = 0; j < 4; ++j) {
        float l = xs[j];
        unsigned u = fkey(l);
        float e = expf(l - m);
        if (lv == 1 && e >= ethr) ++locthr;
        if ((u >> hish) == p) {
          unsigned long long E = (unsigned long long)((double)e * FPSCALE);
          unsigned b = (u >> bsh) & 255u;
          atomicAdd(&cntL[b], 1u);
          atomicAdd(&esumL[b], E);
        }
      }
    }
    for (int i = tail + tid; i < V; i += TB) {
      float l = rp[i];
      unsigned u = fkey(l);
      float e = expf(l - m);
      if (lv == 1 && e >= ethr) ++locthr;
      if ((u >> hish) == p) {
        atomicAdd(&cntL[(u >> bsh) & 255u], 1u);
        atomicAdd(&esumL[(u >> bsh) & 255u],
                  (unsigned long long)((double)e * FPSCALE));
      }
    }
    if (lv == 1) atomicAdd(&s_cthr, locthr);
    __syncthreads();
    if (tid == 0) {
      int fb = -1, lastNE = -1;
      unsigned long long accNE = acc; unsigned CNE = Crun;
      for (int b = 255; b >= 0; --b) {
        unsigned c = cntL[b]; unsigned long long sb = esumL[b];
        if (c && acc + sb > P) { fb = b; break; }
        if (c) { lastNE = b; accNE = acc; CNE = Crun; }
        acc += sb; Crun += c;
      }
      if (fb < 0) { fb = lastNE; acc = accNE; Crun = CNE; }
      pfx = (pfx << 8) | (unsigned)fb;
      if (lv < 3) {
        s_pfx = pfx;
      } else {
        // exact key group: each element contributed an identical integer E
        unsigned c = cntL[fb];
        unsigned long long E =
            (unsigned long long)((double)expf(keyf(pfx) - m) * FPSCALE);
        unsigned q;
        if (E == 0ull) q = c;
        else {
          unsigned long long tmax = (P >= acc) ? (P - acc) / E : 0ull;
          unsigned long long qq = tmax + 1ull;
          q = (qq > (unsigned long long)c) ? c : (unsigned)qq;
        }
        unsigned cp = Crun + q;
        // ---- K* = min(c_p, c_thr, top_k) ----
        int kk = top_ks[row];
        if (kk < 1) kk = 1;
        if (kk > V) kk = V;
        unsigned Kother = ((unsigned)kk < s_cthr) ? (unsigned)kk : s_cthr;
        if (Kother < 1u) Kother = 1u;
        if (cp <= Kother) {
          // top-p binds: its cutoff IS the K*-th key -> skip rank descent
          s_kkey = pfx;
          s_q = q;
          s_needRank = 0;
        } else {
          s_Kstar = Kother;
          s_needRank = 1;
          // rank level-0 walk (cnt0 intact)
          unsigned C2 = 0u; int rb = 0;
          for (int b = 255; b >= 0; --b) {
            if (C2 + cnt0[b] >= Kother) { rb = b; break; }
            C2 += cnt0[b];
          }
          Crun = C2;                       // reuse for rank descent
          s_pfx = (unsigned)rb;
        }
      }
    }
    __syncthreads();
  }

  // ---- scans 5-7: rank descent (only when top_k / threshold binds) ----
  if (s_needRank) {                        // uniform block-wide branch
    for (int lv = 1; lv <= 3; ++lv) {
      if (tid < 256) cntL[tid] = 0u;
      __syncthreads();
      const unsigned p = s_pfx;
      const int hish = 32 - 8 * lv;
      const int bsh = 24 - 8 * lv;
      for (int i = tid; i < nv4; i += TB) {
        __builtin_prefetch(rp4 + i + 4 * TB, 0, 0);
        float4 x = rp4[i];
        float xs[4] = {x.x, x.y, x.z, x.w};
#pragma unroll
        for (int j = 0; j < 4; ++j) {
          unsigned u = fkey(xs[j]);
          if ((u >> hish) == p) atomicAdd(&cntL[(u >> bsh) & 255u], 1u);
        }
      }
      for (int i = tail + tid; i < V; i += TB) {
        unsigned u = fkey(rp[i]);
        if ((u >> hish) == p) atomicAdd(&cntL[(u >> bsh) & 255u], 1u);
      }
      __syncthreads();
      if (tid == 0) {
        const unsigned K = s_Kstar;
        unsigned C2 = Crun; int fb = 0;
        for (int b = 255; b >= 0; --b) {
          if (C2 + cntL[b] >= K) { fb = b; break; }
          C2 += cntL[b];
        }
        Crun = C2;
        unsigned np = (s_pfx << 8) | (unsigned)fb;
        if (lv < 3) s_pfx = np;
        else { s_kkey = np; s_q = K - C2; }   // cutoff key, kept ties (>=1)
      }
      __syncthreads();
    }
  }
  __syncthreads();

  // ---- scan 8: finalize in place (float4).
  // kept = key > k*  ||  (key == k* && tie_idx < q*); masked -> -inf.
  // Each thread owns 4 consecutive indices, so global index order is
  // (thread, slot) lexicographic.  Per-thread tie counts (0..4) are scanned
  // with a wave32 __shfl_up inclusive scan + per-wave LDS exclusive scan.
  const unsigned kkey = s_kkey;
  const unsigned qstar = s_q;
  const unsigned lane = threadIdx.x & 31u;
  const unsigned wid = threadIdx.x >> 5;
  const int CH = TB * 4;
  unsigned base = 0u;

  for (int c0 = 0; c0 < V; c0 += CH) {
    const int bi = c0 + tid * 4;
    float v[4];
    unsigned u[4];
    bool eq[4];
    bool vld[4];
    const bool vec = (bi + 3 < V);
    if (vec) {
      float4 x = *(const float4*)(rp + bi);
      v[0] = x.x; v[1] = x.y; v[2] = x.z; v[3] = x.w;
#pragma unroll
      for (int j = 0; j < 4; ++j) vld[j] = true;
    } else {
#pragma unroll
      for (int j = 0; j < 4; ++j) {
        vld[j] = (bi + j < V);
        v[j] = vld[j] ? rp[bi + j] : 0.0f;
      }
    }
    unsigned cnt = 0u;
#pragma unroll
    for (int j = 0; j < 4; ++j) {
      u[j] = fkey(v[j]);
      eq[j] = vld[j] && (u[j] == kkey);
      cnt += eq[j] ? 1u : 0u;
    }
    // wave32 inclusive scan of per-thread tie counts
    unsigned inc = cnt;
    for (int d = 1; d < 32; d <<= 1) {
      unsigned n = __shfl_up(inc, d, 32);
      if (lane >= (unsigned)d) inc += n;
    }
    if (lane == 31) wtot[wid] = inc;
    __syncthreads();
    if (tid == 0) {
      unsigned s = 0u;
      for (int w = 0; w < 32; ++w) { wexcl[w] = s; s += wtot[w]; }
      s_btot = s;
    }
    __syncthreads();
    unsigned tj = base + wexcl[wid] + (inc - cnt);   // thread's first tie idx
    float r[4];
#pragma unroll
    for (int j = 0; j < 4; ++j) {
      bool kept = (u[j] > kkey) || (eq[j] && tj < qstar);
      r[j] = kept ? v[j] : neg_inf();
      tj += eq[j] ? 1u : 0u;
    }
    if (vec) {
      float4 x; x.x = r[0]; x.y = r[1]; x.z = r[2]; x.w = r[3];
      *(float4*)(rp + bi) = x;
    } else {
#pragma unroll
      for (int j = 0; j < 4; ++j)
        if (vld[j]) rp[bi + j] = r[j];
    }
    base += s_btot;
    __syncthreads();   // protect wtot/wexcl/s_btot for next chunk
  }
}

// ---------------------------------------------------------------------------
extern "C" void kernel_launch(void* const* d_in, const int* in_sizes, int n_in,
                              void* d_out, int out_size, void* d_ws, size_t ws_size,
                              hipStream_t stream) {
  const float* logits  = (const float*)d_in[0];
  const float* pres    = (const float*)d_in[1];
  const float* freq    = (const float*)d_in[2];
  const float* rep     = (const float*)d_in[3];
  const float* temps   = (const float*)d_in[4];
  const float* top_ps  = (const float*)d_in[5];
  const float* top_as  = (const float*)d_in[6];
  const float* min_ps  = (const float*)d_in[7];
  const int*   prompt  = (const int*)d_in[8];
  const int*   outtok  = (const int*)d_in[9];
  const int*   top_ks  = (const int*)d_in[10];
  float* out = (float*)d_out;

  const int B  = in_sizes[1];             // 256
  const int V  = in_sizes[0] / B;         // 128000
  const int LP = in_sizes[8] / B;         // 1024
  const int LO = in_sizes[9] / B;         // 256

  unsigned* rowmax = (unsigned*)d_ws;     // 4 bytes per row

  k0_init<<<(B + 255) / 256, 256, 0, stream>>>(rowmax, B);
  dim3 g1((V + 1023) / 1024, B);
  k1_scale<<<g1, 256, 0, stream>>>(logits, temps, out, rowmax, V);
  k2_penalize<<<B, 256, 0, stream>>>(logits, prompt, outtok, pres, freq, rep,
                                     temps, out, rowmax, V, LP, LO);
  k3_select_finalize<<<B, TB, 0, stream>>>(out, rowmax, top_ps, top_as,
                                           min_ps, top_ks, V);
}